// SparseRoiCut_8358006358563
// MI455X (gfx1250) — compile-verified
//
#include <hip/hip_runtime.h>
#include <hip/hip_bf16.h>
#include <stdint.h>

// SparseRoiCut for MI455X (gfx1250, wave32).
// Memory-bound (~350MB traffic -> ~15us @ 23.3 TB/s HBM). No dense matmul, so
// WMMA is not applicable; the CDNA5-specific paths used:
//   * global_load_async_to_lds_b128 + s_wait_asynccnt (ASYNCcnt copy engine)
//     to stage box descriptors into LDS,
//   * TH=NT (non-temporal) hints on all write-once output streams so the
//     273MB of streaming stores do not evict the L2-resident feature table
//     (51MB) and ballot masks (6.4MB) that later passes re-read.

#define NBOX   256
#define MAXSEL 262144
#define NFEAT  64

typedef float v4f __attribute__((ext_vector_type(4)));

// ---- CDNA5 async global->LDS copy (ASYNCcnt-tracked) ----------------------
__device__ __forceinline__ unsigned lds_off_of(const void* p) {
    // generic pointer to LDS: low 32 bits are the LDS byte offset
    return (unsigned)(uintptr_t)p;
}
__device__ __forceinline__ void async_ld_b128(unsigned lds_off, const void* gaddr) {
    asm volatile("global_load_async_to_lds_b128 %0, %1, off"
                 :: "v"(lds_off), "v"(gaddr)
                 : "memory");
}
__device__ __forceinline__ void wait_async0() {
    asm volatile("s_wait_asynccnt 0" ::: "memory");
}

// ---------------------------------------------------------------------------
__global__ void k_init(int* __restrict__ counts) {
    counts[threadIdx.x] = 0;
}

// Point-major membership test. One thread = one point; loop over all boxes
// held in LDS. Emits is_inside (float 0/1, coalesced, NT stores), per-wave
// ballot masks (for ordered compaction), and per-box counts.
__global__ void k_inside(const int4*  __restrict__ coords4,
                         const float* __restrict__ bbox,    // [NBOX,2,3]
                         const int*   __restrict__ assoc,   // [NBOX]
                         float*       __restrict__ out_ii,  // [NBOX,N]
                         unsigned*    __restrict__ g_mask,  // [NBOX,NW]
                         int*         __restrict__ g_cnt,   // [NBOX]
                         int N, int NW) {
    __shared__ __align__(16) float s_box[NBOX * 6];   // 6144 B
    __shared__ __align__(16) int   s_assoc[NBOX];     // 1024 B
    __shared__ int s_cnt[NBOX];

    const int tid = threadIdx.x;

    // Stage boxes + associations into LDS with the async copy engine.
    {
        const unsigned bl = lds_off_of(s_box);
        const unsigned al = lds_off_of(s_assoc);
        const char* bg = (const char*)bbox;
        const char* ag = (const char*)assoc;
        const int step = blockDim.x * 16;
        for (int off = tid * 16; off < NBOX * 24; off += step)   // 6144 B
            async_ld_b128(bl + off, bg + off);
        for (int off = tid * 16; off < NBOX * 4; off += step)    // 1024 B
            async_ld_b128(al + off, ag + off);
    }
    s_cnt[tid] = 0;
    wait_async0();        // wave-local ASYNCcnt drain
    __syncthreads();      // publish LDS to all waves

    const int p = blockIdx.x * blockDim.x + tid;
    int cs = -1;
    float fx = 0.f, fy = 0.f, fz = 0.f;
    if (p < N) {
        int4 c = coords4[p];
        fx = (float)c.x; fy = (float)c.y; fz = (float)c.z; cs = c.w;
    }
    const int  wv    = (blockIdx.x * blockDim.x + (tid & ~31)) >> 5;
    const bool lane0 = (tid & 31) == 0;
    const bool wvok  = wv < NW;

    for (int b = 0; b < NBOX; ++b) {
        const float sx = s_box[b * 6 + 0], sy = s_box[b * 6 + 1], sz = s_box[b * 6 + 2];
        const float ex = s_box[b * 6 + 3], ey = s_box[b * 6 + 4], ez = s_box[b * 6 + 5];
        const bool inside = (cs == s_assoc[b]) &
                            (fx >= sx) & (fx < ex) &
                            (fy >= sy) & (fy < ey) &
                            (fz >= sz) & (fz < ez);
        if (p < N) {
            // write-once output stream: non-temporal (th:TH_STORE_NT)
            __builtin_nontemporal_store(inside ? 1.0f : 0.0f,
                                        &out_ii[(size_t)b * N + p]);
        }
        const unsigned bal = __builtin_amdgcn_ballot_w32(inside);
        if (lane0 && wvok) {
            g_mask[(size_t)b * NW + wv] = bal;   // re-read by k_compact: RT
            if (bal) atomicAdd(&s_cnt[b], __popc(bal));
        }
    }
    __syncthreads();
    if (s_cnt[tid]) atomicAdd(&g_cnt[tid], s_cnt[tid]);
}

// Exclusive scan of 256 counts (tiny; single thread).
__global__ void k_scan(const int* __restrict__ counts, int* __restrict__ offs) {
    if (threadIdx.x == 0) {
        int run = 0;
        for (int i = 0; i < NBOX; ++i) { offs[i] = run; run += counts[i]; }
        offs[NBOX] = run;
    }
}

// One block per box: ordered compaction of ballot masks -> (slot -> point).
// Reproduces jnp.nonzero row-major (box-major, then point-index) order.
__global__ void k_compact(const unsigned* __restrict__ g_mask,
                          const int*      __restrict__ offs,
                          int*            __restrict__ sel_pt,
                          int*            __restrict__ sel_box,
                          int NW) {
    __shared__ int s_scan[256];
    __shared__ int s_run;
    const int tid  = threadIdx.x;
    const int b    = blockIdx.x;
    const int boff = offs[b];
    if (tid == 0) s_run = 0;
    const unsigned* row = g_mask + (size_t)b * NW;

    for (int base = 0; base < NW; base += 256) {
        const int w = base + tid;
        // masks are consumed exactly once -> non-temporal load
        const unsigned m = (w < NW) ? __builtin_nontemporal_load(&row[w]) : 0u;
        const int c = __popc(m);
        __syncthreads();
        s_scan[tid] = c;
        __syncthreads();
        for (int off = 1; off < 256; off <<= 1) {
            const int v   = s_scan[tid];
            const int add = (tid >= off) ? s_scan[tid - off] : 0;
            __syncthreads();
            s_scan[tid] = v + add;
            __syncthreads();
        }
        const int excl        = s_scan[tid] - c;
        const int chunk_total = s_scan[255];
        int slot = boff + s_run + excl;
        unsigned mm = m;
        const int pbase = w * 32;
        while (mm) {
            const int bit = __ffs((int)mm) - 1;
            mm &= mm - 1u;
            if (slot < MAXSEL) { sel_pt[slot] = pbase + bit; sel_box[slot] = b; }
            ++slot;
        }
        __syncthreads();
        if (tid == 0) s_run += chunk_total;
    }
}

// 16 lanes per slot: float4 feature copy + extended coords; zero-pad rest.
// Output stores are non-temporal (write-once streams); feature reads stay on
// the default RT path so the 51MB table remains L2-resident across the gather.
__global__ void k_gather(const int4* __restrict__ coords4,
                         const v4f*  __restrict__ feat4,    // [N,16] v4f
                         const int*  __restrict__ sel_pt,
                         const int*  __restrict__ sel_box,
                         const int*  __restrict__ offs,
                         v4f*        __restrict__ outC,     // [MAXSEL] v4f
                         v4f*        __restrict__ outF) {   // [MAXSEL,16] v4f
    const int slot = blockIdx.x * 16 + (threadIdx.x >> 4);
    const int lane = threadIdx.x & 15;
    int total = offs[NBOX];
    if (total > MAXSEL) total = MAXSEL;
    if (slot < total) {
        const int pt = sel_pt[slot];
        const int b  = sel_box[slot];
        const v4f f = feat4[(size_t)pt * 16 + lane];
        __builtin_nontemporal_store(f, &outF[(size_t)slot * 16 + lane]);
        if (lane == 0) {
            const int4 c = coords4[pt];
            const v4f e = {(float)c.x, (float)c.y, (float)c.z, (float)b};
            __builtin_nontemporal_store(e, &outC[slot]);
        }
    } else {
        const v4f z = {0.f, 0.f, 0.f, 0.f};
        __builtin_nontemporal_store(z, &outF[(size_t)slot * 16 + lane]);
        if (lane == 0) __builtin_nontemporal_store(z, &outC[slot]);
    }
}

// ---------------------------------------------------------------------------
extern "C" void kernel_launch(void* const* d_in, const int* in_sizes, int n_in,
                              void* d_out, int out_size, void* d_ws, size_t ws_size,
                              hipStream_t stream) {
    const int*   coords   = (const int*)d_in[0];    // [N,4] int32
    const float* features = (const float*)d_in[1];  // [N,64] f32
    const float* bbox     = (const float*)d_in[2];  // [256,2,3] f32
    const int*   assoc    = (const int*)d_in[3];    // [256] int32

    const int N  = in_sizes[0] / 4;
    const int NW = (N + 31) / 32;

    float* out   = (float*)d_out;
    float* outC  = out;                                        // MAXSEL*4
    float* outF  = out + (size_t)MAXSEL * 4;                   // MAXSEL*64
    float* outII = outF + (size_t)MAXSEL * NFEAT;              // NBOX*N

    int* ws_i      = (int*)d_ws;
    int* g_cnt     = ws_i;                    // 256
    int* offs      = ws_i + 256;              // 257
    int* sel_pt    = ws_i + 1024;             // MAXSEL
    int* sel_box   = sel_pt + MAXSEL;         // MAXSEL
    unsigned* g_mask = (unsigned*)(sel_box + MAXSEL);  // NBOX*NW

    k_init<<<1, NBOX, 0, stream>>>(g_cnt);

    const int nblkA = (N + 255) / 256;
    k_inside<<<nblkA, 256, 0, stream>>>((const int4*)coords, bbox, assoc,
                                        outII, g_mask, g_cnt, N, NW);

    k_scan<<<1, 1, 0, stream>>>(g_cnt, offs);

    k_compact<<<NBOX, 256, 0, stream>>>(g_mask, offs, sel_pt, sel_box, NW);

    k_gather<<<MAXSEL / 16, 256, 0, stream>>>((const int4*)coords,
                                              (const v4f*)features,
                                              sel_pt, sel_box, offs,
                                              (v4f*)outC, (v4f*)outF);
}